// SelectiveScan_36009005809749
// MI455X (gfx1250) — compile-verified
//
#include <hip/hip_runtime.h>
#include <math.h>

// Problem constants (match reference)
#define Bb 2
#define Tt 2048
#define Dd 1024
#define Nn 16
#define Rr 64              // DT_RANK
#define BT (Bb * Tt)       // 4096
#define NPROJ 96           // packed projection width: [B_t(16) | C_t(16) | dt_low(64)]
#define NCHUNK 16          // chunks along T
#define CLEN (Tt / NCHUNK) // 128 steps per chunk
#define TS 16              // timesteps staged into LDS per async tile

typedef __attribute__((ext_vector_type(2))) float v2f;
typedef __attribute__((ext_vector_type(8))) float v8f;

// Generic-vector int4, matching the builtin's expected parameter type.
typedef int v4i_t __attribute__((vector_size(16)));
typedef __attribute__((address_space(1))) v4i_t* gptr_v4i;
typedef __attribute__((address_space(3))) v4i_t* lptr_v4i;

// ---------------------------------------------------------------------------
// Async staging helpers (gfx1250 GLOBAL_LOAD_ASYNC_TO_LDS path, ASYNCcnt).
// Each lane stages one 32-float row (128 B) as 8x b128 async transfers.
// Falls back to plain LDS stores if the builtin is unavailable.
// ---------------------------------------------------------------------------
__device__ __forceinline__ void stage_row_async(const float* grow, float* lrow) {
#if __has_builtin(__builtin_amdgcn_global_load_async_to_lds_b128)
#pragma unroll
  for (int j = 0; j < 8; ++j)
    __builtin_amdgcn_global_load_async_to_lds_b128(
        (gptr_v4i)(grow + j * 4),
        (lptr_v4i)(lrow + j * 4),
        0, 0);
#else
#pragma unroll
  for (int k = 0; k < 32; ++k) lrow[k] = grow[k];
#endif
}

__device__ __forceinline__ void wait_async() {
#if __has_builtin(__builtin_amdgcn_s_wait_asynccnt)
  __builtin_amdgcn_s_wait_asynccnt(0);
#elif __has_builtin(__builtin_amdgcn_global_load_async_to_lds_b128)
  asm volatile("s_wait_asynccnt 0" ::: "memory");
#endif
}

// ---------------------------------------------------------------------------
// GEMM1: P96[m, 0:96] = x[m, :] . [W_B; W_C; W_dt1]^T   (M=4096, K=1024, N=96)
// One wave32 per 16x16 tile, V_WMMA_F32_16X16X4_F32, 256 WMMAs per tile.
// ---------------------------------------------------------------------------
__global__ void proj_gemm_kernel(const float* __restrict__ x,
                                 const float* __restrict__ W_B,
                                 const float* __restrict__ W_C,
                                 const float* __restrict__ W_dt1,
                                 float* __restrict__ P96) {  // (BT, 96)
  const int lane  = threadIdx.x & 31;
  const int wave  = (blockIdx.x * blockDim.x + threadIdx.x) >> 5;
  const int NT    = NPROJ / 16;                // 6 n-tiles
  const int mtile = wave / NT;
  const int ntile = wave % NT;
  const int m     = mtile * 16 + (lane & 15);
  const int khalf = lane >> 4;                 // 0 or 1
  const int n     = ntile * 16 + (lane & 15);  // output column 0..95

  const float* arow = x + (size_t)m * Dd;
  const float* wrow;
  if (n < Nn)           wrow = W_B   + (size_t)n * Dd;
  else if (n < 2 * Nn)  wrow = W_C   + (size_t)(n - Nn) * Dd;
  else                  wrow = W_dt1 + (size_t)(n - 2 * Nn) * Dd;

  v8f acc = {};
#pragma unroll 8
  for (int k0 = 0; k0 < Dd; k0 += 4) {
    const int kk = k0 + khalf * 2;
    v2f a = *(const v2f*)(arow + kk);
    v2f b = *(const v2f*)(wrow + kk);
    acc = __builtin_amdgcn_wmma_f32_16x16x4_f32(false, a, false, b,
                                                (short)0, acc, false, false);
  }

#pragma unroll
  for (int r = 0; r < 8; ++r) {
    const int row = mtile * 16 + r + khalf * 8;
    P96[(size_t)row * NPROJ + n] = acc[r];
  }
}

// ---------------------------------------------------------------------------
// GEMM2: dt[m, n] = softplus( P96[m, 32:96] . W_dt2[n, :] + b_dt2[n] )
// ---------------------------------------------------------------------------
__global__ void dt_gemm_kernel(const float* __restrict__ P96,   // (BT,96)
                               const float* __restrict__ W_dt2, // (1024,64)
                               const float* __restrict__ b_dt2, // (1024)
                               float* __restrict__ dt) {        // (BT,1024)
  const int lane  = threadIdx.x & 31;
  const int wave  = (blockIdx.x * blockDim.x + threadIdx.x) >> 5;
  const int mtile = wave >> 6;       // Dd/16 = 64 n-tiles
  const int ntile = wave & 63;
  const int m     = mtile * 16 + (lane & 15);
  const int khalf = lane >> 4;
  const int n     = ntile * 16 + (lane & 15);

  const float* arow = P96   + (size_t)m * NPROJ + 2 * Nn;  // dt_low (64 contiguous)
  const float* wrow = W_dt2 + (size_t)n * Rr;

  v8f acc = {};
#pragma unroll
  for (int k0 = 0; k0 < Rr; k0 += 4) {
    const int kk = k0 + khalf * 2;
    v2f a = *(const v2f*)(arow + kk);
    v2f b = *(const v2f*)(wrow + kk);
    acc = __builtin_amdgcn_wmma_f32_16x16x4_f32(false, a, false, b,
                                                (short)0, acc, false, false);
  }

  const float bias = b_dt2[n];
#pragma unroll
  for (int r = 0; r < 8; ++r) {
    const int row = mtile * 16 + r + khalf * 8;
    const float v = acc[r] + bias;
    const float sp = (v > 20.0f) ? v : log1pf(expf(v));
    dt[(size_t)row * Dd + n] = sp;
  }
}

// ---------------------------------------------------------------------------
// Chunked selective scan (fused discretization + scan + skip), one kernel.
//   Block: 256 threads = 16 d-channels x 16 T-chunks; one batch b, 16 d's.
//   Phase 1: per (d, chunk) affine summary (P = prod A_bar, Q = local scan).
//   Phase 2: 16 threads combine the 16 chunk summaries serially in LDS,
//            storing each chunk's entry state into sQ (reused).
//   Phase 3: re-scan each chunk from its entry state, emit y + D_skip*x.
// B_t/C_t rows are staged into LDS tiles via async-to-LDS DMA: one 128 B row
// per lane per tile, shared by all 16 d-lanes (16x fewer global reads).
// ---------------------------------------------------------------------------
__global__ void scan_kernel(const float* __restrict__ x,      // (B,T,D)
                            const float* __restrict__ dt,     // (B,T,D)
                            const float* __restrict__ P96,    // (B*T,96): Bt|Ct|..
                            const float* __restrict__ log_A,  // (D,N)
                            const float* __restrict__ D_skip, // (D)
                            float* __restrict__ out) {        // (B,T,D)
  const int dl    = threadIdx.x & 15;   // d within the block's 16-wide d group
  const int chunk = threadIdx.x >> 4;   // 0..15
  const int b     = blockIdx.x >> 6;    // Dd/16 = 64 d-groups per batch
  const int dgrp  = blockIdx.x & 63;
  const int d     = dgrp * 16 + dl;
  const int t0    = chunk * CLEN;

  __shared__ float sP[NCHUNK][16][Nn];      // 16 KB
  __shared__ float sQ[NCHUNK][16][Nn];      // 16 KB (later holds entry states)
  __shared__ float sBC[NCHUNK][TS][2 * Nn]; // 32 KB: per-chunk staged [Bt|Ct] rows

  float A[Nn];
#pragma unroll
  for (int nI = 0; nI < Nn; ++nI) A[nI] = -expf(log_A[d * Nn + nI]);

  const float* xb  = x   + (size_t)b * Tt * Dd;
  const float* dtb = dt  + (size_t)b * Tt * Dd;
  const float* pb  = P96 + (size_t)b * Tt * NPROJ;
  float* ob        = out + (size_t)b * Tt * Dd;

  // ---- Phase 1: chunk summaries (P, Q) ----
  float P[Nn], Q[Nn];
#pragma unroll
  for (int nI = 0; nI < Nn; ++nI) { P[nI] = 1.0f; Q[nI] = 0.0f; }

  for (int tile = 0; tile < CLEN / TS; ++tile) {
    const int tb = t0 + tile * TS;
    __syncthreads();  // previous tile's LDS reads complete before overwrite
    stage_row_async(pb + (size_t)(tb + dl) * NPROJ, &sBC[chunk][dl][0]);
    wait_async();
    __syncthreads();  // staged data visible (also covers fallback path)

    for (int i = 0; i < TS; ++i) {
      const int t = tb + i;
      if (i + 4 < TS) {
        __builtin_prefetch(&xb[(size_t)(t + 4) * Dd + d], 0, 1);
        __builtin_prefetch(&dtb[(size_t)(t + 4) * Dd + d], 0, 1);
      }
      const float xv  = xb[(size_t)t * Dd + d];
      const float dtv = dtb[(size_t)t * Dd + d];
#pragma unroll
      for (int nI = 0; nI < Nn; ++nI) {
        const float Ab   = expf(dtv * A[nI]);
        const float dBn  = (Ab - 1.0f) / (A[nI] + 1e-8f);
        const float Bbar = dBn * xv * sBC[chunk][i][nI];
        Q[nI] = Ab * Q[nI] + Bbar;
        P[nI] *= Ab;
      }
    }
  }
#pragma unroll
  for (int nI = 0; nI < Nn; ++nI) {
    sP[chunk][dl][nI] = P[nI];
    sQ[chunk][dl][nI] = Q[nI];
  }
  __syncthreads();

  // ---- Phase 2: serial combine across 16 chunks (16 threads) ----
  // After this, sQ[c][d][n] holds the ENTRY state for chunk c.
  if (threadIdx.x < 16) {
    const int dd = threadIdx.x;
    float H[Nn];
#pragma unroll
    for (int nI = 0; nI < Nn; ++nI) H[nI] = 0.0f;
    for (int c = 0; c < NCHUNK; ++c) {
#pragma unroll
      for (int nI = 0; nI < Nn; ++nI) {
        const float hprev = H[nI];
        H[nI] = sP[c][dd][nI] * hprev + sQ[c][dd][nI];
        sQ[c][dd][nI] = hprev;  // entry state for chunk c
      }
    }
  }
  __syncthreads();

  // ---- Phase 3: re-scan chunk from entry state, emit outputs ----
  float h[Nn];
#pragma unroll
  for (int nI = 0; nI < Nn; ++nI) h[nI] = sQ[chunk][dl][nI];
  const float dsk = D_skip[d];

  for (int tile = 0; tile < CLEN / TS; ++tile) {
    const int tb = t0 + tile * TS;
    __syncthreads();
    stage_row_async(pb + (size_t)(tb + dl) * NPROJ, &sBC[chunk][dl][0]);
    wait_async();
    __syncthreads();

    for (int i = 0; i < TS; ++i) {
      const int t = tb + i;
      if (i + 4 < TS) {
        __builtin_prefetch(&xb[(size_t)(t + 4) * Dd + d], 0, 1);
        __builtin_prefetch(&dtb[(size_t)(t + 4) * Dd + d], 0, 1);
      }
      const float xv  = xb[(size_t)t * Dd + d];
      const float dtv = dtb[(size_t)t * Dd + d];
      float y = 0.0f;
#pragma unroll
      for (int nI = 0; nI < Nn; ++nI) {
        const float Ab   = expf(dtv * A[nI]);
        const float dBn  = (Ab - 1.0f) / (A[nI] + 1e-8f);
        const float Bbar = dBn * xv * sBC[chunk][i][nI];
        h[nI] = Ab * h[nI] + Bbar;
        y += h[nI] * sBC[chunk][i][Nn + nI];
      }
      ob[(size_t)t * Dd + d] = y + dsk * xv;
    }
  }
}

// ---------------------------------------------------------------------------
extern "C" void kernel_launch(void* const* d_in, const int* in_sizes, int n_in,
                              void* d_out, int out_size, void* d_ws, size_t ws_size,
                              hipStream_t stream) {
  const float* x      = (const float*)d_in[0];
  const float* W_B    = (const float*)d_in[1];
  const float* W_C    = (const float*)d_in[2];
  const float* W_dt1  = (const float*)d_in[3];
  const float* W_dt2  = (const float*)d_in[4];
  const float* b_dt2  = (const float*)d_in[5];
  const float* log_A  = (const float*)d_in[6];
  const float* D_skip = (const float*)d_in[7];
  float* out = (float*)d_out;

  // Workspace layout (floats): dt (BT*Dd) | P96 (BT*96)
  float* ws  = (float*)d_ws;
  float* dt  = ws;                          // 4,194,304 floats
  float* P96 = dt + (size_t)BT * Dd;        //   393,216 floats

  // GEMM1: 256 m-tiles * 6 n-tiles = 1536 waves -> 192 blocks * 256 threads
  proj_gemm_kernel<<<192, 256, 0, stream>>>(x, W_B, W_C, W_dt1, P96);

  // GEMM2: 256 m-tiles * 64 n-tiles = 16384 waves -> 2048 blocks * 256 threads
  dt_gemm_kernel<<<2048, 256, 0, stream>>>(P96, W_dt2, b_dt2, dt);

  // Chunked scan: grid = B * (D/16) = 128 blocks of 256 (16 d x 16 chunks)
  scan_kernel<<<Bb * (Dd / 16), 256, 0, stream>>>(x, dt, P96, log_A, D_skip, out);
}